// TPUMoELayer_19756849562325
// MI455X (gfx1250) — compile-verified
//
#include <hip/hip_runtime.h>
#include <math.h>

// ---- problem constants (fixed by the reference) ----
#define T_TOK 8192   // B*S
#define DD    1024
#define EE    8
#define HH    4096
#define KTOP  2

typedef float v2f __attribute__((ext_vector_type(2)));
typedef float v8f __attribute__((ext_vector_type(8)));

__device__ __forceinline__ v8f wmma_f32(v2f a, v2f b, v8f c) {
  // V_WMMA_F32_16X16X4_F32 : D = A(16x4) * B(4x16) + C(16x16), fp32 throughout
  return __builtin_amdgcn_wmma_f32_16x16x4_f32(
      /*neg_a=*/false, a, /*neg_b=*/false, b,
      /*c_mod=*/(short)0, c, /*reuse_a=*/false, /*reuse_b=*/false);
}

__device__ __forceinline__ float gelu_tanh(float x) {
  // jax.nn.gelu default (approximate=True)
  float x3 = x * x * x;
  return 0.5f * x * (1.0f + tanhf(0.7978845608028654f * (x + 0.044715f * x3)));
}

// ------------------------------------------------------------------
// kernel 0: zero the 2*E routing counters (deterministic every call)
// ------------------------------------------------------------------
__global__ void zero_counts_kernel(int* __restrict__ counts) {
  if (threadIdx.x < 2 * EE) counts[threadIdx.x] = 0;
}

// ------------------------------------------------------------------
// kernel 1: router. one wave32 per token.
//   logits = x_t . Wr ; softmax ; top-2 ; append to (expert, k) lists
// ------------------------------------------------------------------
__global__ __launch_bounds__(256) void router_kernel(
    const float* __restrict__ x, const float* __restrict__ Wr,
    int* __restrict__ counts, int* __restrict__ tokL, float* __restrict__ wgtL) {
  const int wave = threadIdx.x >> 5;
  const int lane = threadIdx.x & 31;
  const int t = blockIdx.x * 8 + wave;

  float p[EE];
#pragma unroll
  for (int e = 0; e < EE; ++e) p[e] = 0.f;

  const float* xr = x + (size_t)t * DD;
  for (int d = lane; d < DD; d += 32) {
    float xv = xr[d];
    const float* wr = Wr + d * EE;  // 8 consecutive floats per row
#pragma unroll
    for (int e = 0; e < EE; ++e) p[e] += xv * wr[e];
  }
  // full wave32 butterfly reduction
#pragma unroll
  for (int off = 16; off > 0; off >>= 1) {
#pragma unroll
    for (int e = 0; e < EE; ++e) p[e] += __shfl_xor(p[e], off, 32);
  }

  if (lane == 0) {
    float mx = p[0];
#pragma unroll
    for (int e = 1; e < EE; ++e) mx = fmaxf(mx, p[e]);
    float pr[EE];
    float s = 0.f;
#pragma unroll
    for (int e = 0; e < EE; ++e) { pr[e] = __expf(p[e] - mx); s += pr[e]; }
    float inv = 1.0f / s;
    // top-1 (first occurrence on ties, matching lax.top_k)
    int i0 = 0;
#pragma unroll
    for (int e = 1; e < EE; ++e) if (p[e] > p[i0]) i0 = e;
    // top-2
    int i1 = (i0 == 0) ? 1 : 0;
#pragma unroll
    for (int e = 0; e < EE; ++e) if (e != i0 && p[e] > p[i1]) i1 = e;

    int lid0 = i0 * 2 + 0;
    int pos0 = atomicAdd(&counts[lid0], 1);
    tokL[(size_t)lid0 * T_TOK + pos0] = t;
    wgtL[(size_t)lid0 * T_TOK + pos0] = pr[i0] * inv;

    int lid1 = i1 * 2 + 1;
    int pos1 = atomicAdd(&counts[lid1], 1);
    tokL[(size_t)lid1 * T_TOK + pos1] = t;
    wgtL[(size_t)lid1 * T_TOK + pos1] = pr[i1] * inv;
  }
}

// ------------------------------------------------------------------
// kernel 2: fused expert FFN over a 16-row gathered tile.
//   out_tile = w_row * ( gelu(X*W1 + b1) * W2 + b2 )
// Launched twice: kpass=0 -> "out =", kpass=1 -> "out +=".
// ------------------------------------------------------------------
#define XPITCH 516   // 512 + 4 : 16 rows hit distinct LDS bank pairs
#define HPITCH 132   // 128 + 4

__global__ __launch_bounds__(256) void moe_ffn_kernel(
    const float* __restrict__ x, const float* __restrict__ W1,
    const float* __restrict__ b1, const float* __restrict__ W2,
    const float* __restrict__ b2, const int* __restrict__ counts,
    const int* __restrict__ tokL, const float* __restrict__ wgtL,
    float* __restrict__ out, int kpass) {
  const int e = blockIdx.y;
  const int lid = e * 2 + kpass;
  const int cnt = counts[lid];
  const int row0 = blockIdx.x * 16;
  if (row0 >= cnt) return;  // block-uniform exit

  __shared__ float Xs[16 * XPITCH];  // 33.0 KB : half of the X tile (512 cols)
  __shared__ float Hs[16 * HPITCH];  //  8.4 KB : gelu(h) chunk (128 cols)
  __shared__ int   tokS[16];
  __shared__ float wS[16];

  const int tid  = threadIdx.x;
  const int wv   = tid >> 5;        // 8 waves
  const int lane = tid & 31;
  const int half = lane >> 4;       // 0 | 1
  const int l16  = lane & 15;

  if (tid < 16) {
    int r = row0 + tid;
    int ok = (r < cnt);
    tokS[tid] = ok ? tokL[(size_t)lid * T_TOK + r] : 0;  // safe row for pads
    wS[tid]   = ok ? wgtL[(size_t)lid * T_TOK + r] : 0.f;
  }

  const float* W1e = W1 + (size_t)e * DD * HH;
  const float* W2e = W2 + (size_t)e * HH * DD;

  v8f acc[8];                       // wave's 16 x 128 slice of out tile
#pragma unroll
  for (int j = 0; j < 8; ++j) acc[j] = (v8f){0.f, 0.f, 0.f, 0.f, 0.f, 0.f, 0.f, 0.f};

  const int sm = tid >> 4;          // staging row
  const int sc = (tid & 15) * 32;   // staging col chunk inside a 512-half

  for (int hc = 0; hc < HH / 128; ++hc) {
    const int hbase = hc * 128 + wv * 16;      // this wave's 16 h-columns
    if (hc + 1 < HH / 128)
      __builtin_prefetch(W2e + (size_t)(hc + 1) * 128 * DD + wv * 128 + l16, 0, 3);

    // ---------------- phase A: h = gelu(X * W1 + b1), 16 x 128 ----------------
    v8f hacc = (v8f){0.f, 0.f, 0.f, 0.f, 0.f, 0.f, 0.f, 0.f};
    const float* w1p = W1e + hbase + l16;
#pragma unroll
    for (int dh = 0; dh < 2; ++dh) {
      __syncthreads();  // everyone done reading previous Xs/Hs contents
      {   // stage 512-column half of the gathered X tile
        const float4* src = (const float4*)(x + (size_t)tokS[sm] * DD + dh * 512 + sc);
        float4* dst = (float4*)(Xs + sm * XPITCH + sc);
#pragma unroll
        for (int i = 0; i < 8; ++i) dst[i] = src[i];
      }
      __syncthreads();
      const float* xrow = Xs + l16 * XPITCH + 2 * half;  // A row = l16
      const int kg = dh * 512;
#pragma unroll 4
      for (int kk = 0; kk < 512; kk += 4) {
        v2f a = *(const v2f*)(xrow + kk);                 // ds_load_b64
        v2f b;
        b.x = w1p[(size_t)(kg + kk + 2 * half) * HH];
        b.y = w1p[(size_t)(kg + kk + 2 * half + 1) * HH];
        hacc = wmma_f32(a, b, hacc);
      }
    }
    // bias + gelu, write 16x16 tile into Hs at columns [wv*16, wv*16+16)
    {
      float b1v = b1[(size_t)e * HH + hbase + l16];
      float* hdst = Hs + (8 * half) * HPITCH + wv * 16 + l16;
#pragma unroll
      for (int r = 0; r < 8; ++r)
        hdst[r * HPITCH] = gelu_tanh(hacc[r] + b1v);
    }
    __syncthreads();

    // ---------------- phase B: acc += h(16x128) * W2_chunk(128x1024) ----------
    const float* w2p = W2e + (size_t)hc * 128 * DD + wv * 128 + l16;
    const float* hrow = Hs + l16 * HPITCH + 2 * half;
#pragma unroll 2
    for (int kk = 0; kk < 128; kk += 4) {
      v2f a = *(const v2f*)(hrow + kk);                   // shared across 8 tiles
      size_t r0 = (size_t)(kk + 2 * half) * DD;
      size_t r1 = r0 + DD;
#pragma unroll
      for (int j = 0; j < 8; ++j) {
        v2f b;
        b.x = w2p[r0 + j * 16];
        b.y = w2p[r1 + j * 16];
        acc[j] = wmma_f32(a, b, acc[j]);
      }
    }
  }

  // ---------------- epilogue: weight, bias2, scatter --------------------------
#pragma unroll
  for (int j = 0; j < 8; ++j) {
    int n = wv * 128 + j * 16 + l16;
    float b2v = b2[(size_t)e * DD + n];
#pragma unroll
    for (int r = 0; r < 8; ++r) {
      int m = r + 8 * half;
      if (row0 + m < cnt) {
        float val = (acc[j][r] + b2v) * wS[m];
        float* po = out + (size_t)tokS[m] * DD + n;
        if (kpass == 0) *po = val;   // k=0 pass fully initializes out
        else            *po += val;  // k=1 pass accumulates (ordered by launch)
      }
    }
  }
}

// ------------------------------------------------------------------
extern "C" void kernel_launch(void* const* d_in, const int* in_sizes, int n_in,
                              void* d_out, int out_size, void* d_ws, size_t ws_size,
                              hipStream_t stream) {
  (void)in_sizes; (void)n_in; (void)out_size; (void)ws_size;
  const float* x  = (const float*)d_in[0];
  const float* Wr = (const float*)d_in[1];
  const float* W1 = (const float*)d_in[2];
  const float* b1 = (const float*)d_in[3];
  const float* W2 = (const float*)d_in[4];
  const float* b2 = (const float*)d_in[5];
  float* out = (float*)d_out;

  // workspace layout: counts[16] | tokL[16*T] | wgtL[16*T]  (~1.05 MB)
  char* ws = (char*)d_ws;
  int*   counts = (int*)ws;
  int*   tokL   = (int*)(ws + 256);
  float* wgtL   = (float*)(ws + 256 + (size_t)2 * EE * T_TOK * sizeof(int));

  zero_counts_kernel<<<1, 32, 0, stream>>>(counts);
  router_kernel<<<T_TOK / 8, 256, 0, stream>>>(x, Wr, counts, tokL, wgtL);

  dim3 grid(T_TOK / 16, EE);
  moe_ffn_kernel<<<grid, 256, 0, stream>>>(x, W1, b1, W2, b2, counts, tokL, wgtL, out, 0);
  moe_ffn_kernel<<<grid, 256, 0, stream>>>(x, W1, b1, W2, b2, counts, tokL, wgtL, out, 1);
}